// ClusterPooling_21406117003594
// MI455X (gfx1250) — compile-verified
//
#include <hip/hip_runtime.h>

// Float4 vector type for 128-bit global accesses (global_load_b128 / b128 stores).
typedef float f4 __attribute__((ext_vector_type(4)));

// ---------------------------------------------------------------------------
// Scatter-accumulate, specialized for D=256 (D4=64, SHIFT=6). One thread per
// (node, 4-feature chunk); gid = node*64 + chunk. Placed FIRST in the file so
// the disasm snippet reaches the atomic ops: expecting global_atomic_add_f32
// (hardware f32 atomic, no CAS loop).
//
// Wave32: a wave spans 32 consecutive gids and 64 chunks per node, so `node`
// is wave-uniform -> readfirstlane keeps it (and the cluster id) scalar.
// x/h are streamed exactly once -> non-temporal b128 loads keep the 2x205 MB
// streams from evicting the L2-resident 102 MB accumulator region that the
// atomic RMW traffic wants hot in the 192 MB L2.
// ---------------------------------------------------------------------------
__global__ void scatter_kernel_d256(const f4* __restrict__ x,
                                    const f4* __restrict__ h,
                                    const int* __restrict__ cmap,
                                    float* __restrict__ xsum,
                                    float* __restrict__ hsum,
                                    float* __restrict__ counts,
                                    int total) {
    int gid = blockIdx.x * blockDim.x + threadIdx.x;
    if (gid >= total) return;

    int node  = __builtin_amdgcn_readfirstlane(gid >> 6);  // wave-uniform
    int chunk = gid & 63;
    int c = cmap[node];

    f4 xv = __builtin_nontemporal_load(&x[gid]);
    f4 hv = __builtin_nontemporal_load(&h[gid]);

    float* xd = xsum + ((size_t)c << 8) + (chunk << 2);
    float* hd = hsum + ((size_t)c << 8) + (chunk << 2);

    // Hardware f32 atomics (global_atomic_add_f32), no CAS loop.
    unsafeAtomicAdd(xd + 0, xv.x);
    unsafeAtomicAdd(xd + 1, xv.y);
    unsafeAtomicAdd(xd + 2, xv.z);
    unsafeAtomicAdd(xd + 3, xv.w);
    unsafeAtomicAdd(hd + 0, hv.x);
    unsafeAtomicAdd(hd + 1, hv.y);
    unsafeAtomicAdd(hd + 2, hv.z);
    unsafeAtomicAdd(hd + 3, hv.w);

    if (chunk == 0) {
        unsafeAtomicAdd(&counts[c], 1.0f);
    }
}

// ---------------------------------------------------------------------------
// Generic-shape fallback (any power-of-two D4).
// ---------------------------------------------------------------------------
__global__ void scatter_kernel(const f4* __restrict__ x,
                               const f4* __restrict__ h,
                               const int* __restrict__ cmap,
                               float* __restrict__ xsum,
                               float* __restrict__ hsum,
                               float* __restrict__ counts,
                               int total, int D, int shift) {
    int gid = blockIdx.x * blockDim.x + threadIdx.x;
    if (gid >= total) return;

    const int mask = (1 << shift) - 1;
    int node  = gid >> shift;
    int chunk = gid & mask;
    if (shift >= 5) {
        node = __builtin_amdgcn_readfirstlane(node);  // wave-uniform
    }
    int c = cmap[node];

    f4 xv = __builtin_nontemporal_load(&x[gid]);
    f4 hv = __builtin_nontemporal_load(&h[gid]);

    float* xd = xsum + (size_t)c * D + (chunk << 2);
    float* hd = hsum + (size_t)c * D + (chunk << 2);

    unsafeAtomicAdd(xd + 0, xv.x);
    unsafeAtomicAdd(xd + 1, xv.y);
    unsafeAtomicAdd(xd + 2, xv.z);
    unsafeAtomicAdd(xd + 3, xv.w);
    unsafeAtomicAdd(hd + 0, hv.x);
    unsafeAtomicAdd(hd + 1, hv.y);
    unsafeAtomicAdd(hd + 2, hv.z);
    unsafeAtomicAdd(hd + 3, hv.w);

    if (chunk == 0) {
        unsafeAtomicAdd(&counts[c], 1.0f);
    }
}

// ---------------------------------------------------------------------------
// Zero kernels: re-initialize accumulators every call (harness poisons d_out /
// d_ws with 0xAA once and never re-poisons between replays).
// ---------------------------------------------------------------------------
__global__ void zero_f4_kernel(f4* __restrict__ p, int n4) {
    int i      = blockIdx.x * blockDim.x + threadIdx.x;
    int stride = gridDim.x * blockDim.x;
    f4 z = {0.f, 0.f, 0.f, 0.f};
    for (; i < n4; i += stride) p[i] = z;
}

__global__ void zero_f1_kernel(float* __restrict__ p, int n) {
    int i      = blockIdx.x * blockDim.x + threadIdx.x;
    int stride = gridDim.x * blockDim.x;
    for (; i < n; i += stride) p[i] = 0.f;
}

// ---------------------------------------------------------------------------
// Finalize: sums -> means. One thread per (cluster, 4-feature chunk).
// Final results are written once and never re-read -> non-temporal stores.
// ---------------------------------------------------------------------------
__global__ void finalize_kernel(f4* __restrict__ xsum,
                                f4* __restrict__ hsum,
                                const float* __restrict__ counts,
                                int total4, int shift) {
    int i = blockIdx.x * blockDim.x + threadIdx.x;
    if (i >= total4) return;

    int c = i >> shift;
    if (shift >= 5) {
        c = __builtin_amdgcn_readfirstlane(c);  // wave-uniform cluster id
    }
    float cnt = counts[c];
    float inv = 1.0f / fmaxf(cnt, 1.0f);

    f4 xv = xsum[i];
    f4 hv = hsum[i];
    __builtin_nontemporal_store(xv * inv, &xsum[i]);
    __builtin_nontemporal_store(hv * inv, &hsum[i]);
}

// ---------------------------------------------------------------------------
// pos_sampled = pos[sample_index]  ([C,3] gather; tiny, latency-bound)
// ---------------------------------------------------------------------------
__global__ void gather_pos_kernel(const float* __restrict__ pos,
                                  const int* __restrict__ sidx,
                                  float* __restrict__ out, int C) {
    int i = blockIdx.x * blockDim.x + threadIdx.x;
    if (i >= C) return;
    int s = sidx[i];
    out[3 * i + 0] = pos[3 * s + 0];
    out[3 * i + 1] = pos[3 * s + 1];
    out[3 * i + 2] = pos[3 * s + 2];
}

extern "C" void kernel_launch(void* const* d_in, const int* in_sizes, int n_in,
                              void* d_out, int out_size, void* d_ws, size_t ws_size,
                              hipStream_t stream) {
    const float* x    = (const float*)d_in[0];  // [N, D]
    const float* h    = (const float*)d_in[1];  // [N, D]
    const float* pos  = (const float*)d_in[2];  // [N, 3]
    const int*   cmap = (const int*)d_in[3];    // [N]
    const int*   sidx = (const int*)d_in[4];    // [C]

    const int N  = in_sizes[3];        // 200000
    const int C  = in_sizes[4];        // 50000
    const int D  = in_sizes[0] / N;    // 256
    const int D4 = D >> 2;             // 64 (power of two)
    int shift = 0;
    while ((1 << shift) < D4) ++shift; // log2(D4) = 6

    float* xsum   = (float*)d_out;                 // [C, D]
    float* hsum   = xsum + (size_t)C * D;          // [C, D]
    float* posout = hsum + (size_t)C * D;          // [C, 3]
    float* counts = (float*)d_ws;                  // [C] floats in scratch

    // 1) Zero accumulators + counts (every call).
    const int n4sums = (2 * C * D) >> 2;           // 6.4M float4
    zero_f4_kernel<<<4096, 256, 0, stream>>>((f4*)xsum, n4sums);
    zero_f1_kernel<<<64,   256, 0, stream>>>(counts, C);

    // 2) Scatter-add features + counts.
    const int total = N * D4;                      // 12.8M threads
    if (D == 256) {
        scatter_kernel_d256<<<(total + 255) / 256, 256, 0, stream>>>(
            (const f4*)x, (const f4*)h, cmap, xsum, hsum, counts, total);
    } else {
        scatter_kernel<<<(total + 255) / 256, 256, 0, stream>>>(
            (const f4*)x, (const f4*)h, cmap, xsum, hsum, counts, total, D, shift);
    }

    // 3) Divide by counts (guarding empty clusters).
    const int total4 = C * D4;                     // 3.2M threads
    finalize_kernel<<<(total4 + 255) / 256, 256, 0, stream>>>(
        (f4*)xsum, (f4*)hsum, counts, total4, shift);

    // 4) Position gather.
    gather_pos_kernel<<<(C + 255) / 256, 256, 0, stream>>>(pos, sidx, posout, C);
}